// TsallisSecant_57329223467528
// MI455X (gfx1250) — compile-verified
//
#include <hip/hip_runtime.h>
#include <hip/hip_bf16.h>
#include <math.h>

// Tsallis alpha-entmax (alpha = 1.5) secant solver, one workgroup per row.
// Row (128 KB) staged once into LDS via CDNA5 async global->LDS copies, all
// secant passes run out of LDS. Row padded to 32768 elements with -inf so
// every hot pass is a fixed 16-vec4-per-thread fully-unrolled loop; -inf pad
// contributes exactly 0 to every f(tau) sum and never wins the max.
// 512 threads (16 wave32) per row: halves the serial per-iteration pass
// latency vs 256, and LDS caps occupancy at 2 WGs/WGP either way.

#define TS_D      32000
#define TS_D4     (TS_D / 4)        // 8000 real float4 per row
#define TS_DP     32768             // padded element count
#define TS_DP4    (TS_DP / 4)       // 8192 float4
#define TS_BLOCK  512               // 16 wave32 waves
#define TS_NWAVE  (TS_BLOCK / 32)
#define TS_VPT    (TS_DP4 / TS_BLOCK)  // 16 vec4 per thread, exact
#define TS_NITER  100
#define TS_TOL    1e-5f
#define TS_GPZERO (-2.0f)
// gp(1/d, alpha=1.5) = 2*(sqrt(1/32000) - 1), computed in double, cast to f32
#define TS_GP_HI  (-1.98881966011250105e0f)

// GCC-style int4 vector (matches the async builtin's expected pointee type)
typedef int ts_v4i __attribute__((vector_size(16)));
typedef __attribute__((address_space(1))) ts_v4i ts_gv4i;  // global
typedef __attribute__((address_space(3))) ts_v4i ts_sv4i;  // LDS

__device__ __forceinline__ float ts_wred_sum(float v) {
#pragma unroll
  for (int off = 16; off > 0; off >>= 1) v += __shfl_xor(v, off, 32);
  return v;
}

__device__ __forceinline__ float ts_wred_max(float v) {
#pragma unroll
  for (int off = 16; off > 0; off >>= 1) v = fmaxf(v, __shfl_xor(v, off, 32));
  return v;
}

// Deterministic block reduction: fixed shuffle tree + sequential combine of
// the 16 wave partials by thread 0, broadcast through LDS.
__device__ __forceinline__ float ts_bred_sum(float v, float* red, float* bc) {
  v = ts_wred_sum(v);
  const int wid = threadIdx.x >> 5;
  if ((threadIdx.x & 31) == 0) red[wid] = v;
  __syncthreads();
  if (threadIdx.x == 0) {
    float s = red[0];
#pragma unroll
    for (int w = 1; w < TS_NWAVE; ++w) s += red[w];
    *bc = s;
  }
  __syncthreads();
  return *bc;
}

__device__ __forceinline__ float ts_bred_max(float v, float* red, float* bc) {
  v = ts_wred_max(v);
  const int wid = threadIdx.x >> 5;
  if ((threadIdx.x & 31) == 0) red[wid] = v;
  __syncthreads();
  if (threadIdx.x == 0) {
    float s = red[0];
#pragma unroll
    for (int w = 1; w < TS_NWAVE; ++w) s = fmaxf(s, red[w]);
    *bc = s;
  }
  __syncthreads();
  return *bc;
}

// p(x - tau) = (1 + 0.5*max(x - tau, -2))^2 ; fma(0.5,u,1) == (1 + 0.5*u)
// exactly since 0.5*u is exact in binary fp.
__device__ __forceinline__ float ts_pacc(float x, float tau, float acc) {
  float u = fmaxf(x - tau, TS_GPZERO);
  float p = fmaf(0.5f, u, 1.0f);
  return fmaf(p, p, acc);
}

__device__ __forceinline__ float ts_pval(float x, float tau) {
  float u = fmaxf(x - tau, TS_GPZERO);
  float p = fmaf(0.5f, u, 1.0f);
  return p * p;
}

__global__ __launch_bounds__(TS_BLOCK) void
tsallis_secant_kernel(const float* __restrict__ X, float* __restrict__ Out) {
  __shared__ float xs[TS_DP];       // ~128.8 KB row tile (CDNA5: 320 KB/WG)
  __shared__ float red[TS_NWAVE];
  __shared__ float bc;

  const int tid = threadIdx.x;
  const size_t rowoff = (size_t)blockIdx.x * (size_t)TS_D;
  const float4* __restrict__ xg4 = (const float4*)(X + rowoff);
  float4* __restrict__ og4 = (float4*)(Out + rowoff);
  float4* xs4 = (float4*)xs;

  // ---- Stage row into LDS via async global->LDS DMA path (ASYNCcnt) ----
#if defined(__gfx1250__) && __has_builtin(__builtin_amdgcn_global_load_async_to_lds_b128)
  for (int i = tid; i < TS_D4; i += TS_BLOCK) {
    __builtin_amdgcn_global_load_async_to_lds_b128(
        (ts_gv4i*)(xg4 + i), (ts_sv4i*)(xs4 + i), 0, 0);
  }
#else
  for (int i = tid; i < TS_D4; i += TS_BLOCK) xs4[i] = xg4[i];
#endif
  // Fill pad [32000, 32768) with -inf: contributes exactly 0 to every sum
  // (clamps to -2 -> p == 0) and never wins the row max.
  if (tid < TS_DP4 - TS_D4) {
    const float ninf = -__builtin_inff();
    float4 f; f.x = ninf; f.y = ninf; f.z = ninf; f.w = ninf;
    xs4[TS_D4 + tid] = f;
  }
#if defined(__gfx1250__) && __has_builtin(__builtin_amdgcn_global_load_async_to_lds_b128)
#if __has_builtin(__builtin_amdgcn_s_wait_asynccnt)
  __builtin_amdgcn_s_wait_asynccnt(0);
#else
  asm volatile("s_wait_asynccnt 0" ::: "memory");
#endif
#endif
  __syncthreads();

  // ---- Row max (fixed-trip unrolled pass over padded row) ----
  float m = -__builtin_inff();
#pragma unroll
  for (int k = 0; k < TS_VPT; ++k) {
    float4 v = xs4[tid + k * TS_BLOCK];
    m = fmaxf(m, fmaxf(fmaxf(v.x, v.y), fmaxf(v.z, v.w)));
  }
  const float max_val = ts_bred_max(m, red, &bc);

  float lo = max_val;               // gp(1.0) == 0
  float hi = max_val - TS_GP_HI;

  // ---- Initial f_lo, f_hi in one pass ----
  float a_lo = 0.0f, a_hi = 0.0f;
#pragma unroll
  for (int k = 0; k < TS_VPT; ++k) {
    float4 v = xs4[tid + k * TS_BLOCK];
    a_lo = ts_pacc(v.x, lo, a_lo); a_lo = ts_pacc(v.y, lo, a_lo);
    a_lo = ts_pacc(v.z, lo, a_lo); a_lo = ts_pacc(v.w, lo, a_lo);
    a_hi = ts_pacc(v.x, hi, a_hi); a_hi = ts_pacc(v.y, hi, a_hi);
    a_hi = ts_pacc(v.z, hi, a_hi); a_hi = ts_pacc(v.w, hi, a_hi);
  }
  float f_lo = ts_bred_sum(a_lo, red, &bc) - 1.0f;
  float f_hi = ts_bred_sum(a_hi, red, &bc) - 1.0f;

  // ---- Secant iterations (workgroup-uniform early exit: once the mask is
  // false it stays false and the carry is a fixed point in the reference) ----
  for (int it = 0; it < TS_NITER; ++it) {
    const float diff = f_lo - f_hi;
    if (diff * diff < TS_TOL) break;
    const float tau = (lo * f_hi - hi * f_lo) / (f_hi - f_lo);
    lo = hi; f_lo = f_hi; hi = tau;
    float acc = 0.0f;
#pragma unroll
    for (int k = 0; k < TS_VPT; ++k) {
      float4 v = xs4[tid + k * TS_BLOCK];
      acc = ts_pacc(v.x, hi, acc);
      acc = ts_pacc(v.y, hi, acc);
      acc = ts_pacc(v.z, hi, acc);
      acc = ts_pacc(v.w, hi, acc);
    }
    f_hi = ts_bred_sum(acc, red, &bc) - 1.0f;
  }

  // ---- Output: final p_hi == p(X - hi_final); real elements only ----
  for (int i = tid; i < TS_D4; i += TS_BLOCK) {
    float4 v = xs4[i];
    float4 o;
    o.x = ts_pval(v.x, hi);
    o.y = ts_pval(v.y, hi);
    o.z = ts_pval(v.z, hi);
    o.w = ts_pval(v.w, hi);
    og4[i] = o;
  }
}

extern "C" void kernel_launch(void* const* d_in, const int* in_sizes, int n_in,
                              void* d_out, int out_size, void* d_ws, size_t ws_size,
                              hipStream_t stream) {
  const float* X = (const float*)d_in[0];
  float* Out = (float*)d_out;
  const int rows = in_sizes[0] / TS_D;   // 4096 x 32000 per the reference
  tsallis_secant_kernel<<<dim3(rows), dim3(TS_BLOCK), 0, stream>>>(X, Out);
}